// GetCostVolume_68719476942
// MI455X (gfx1250) — compile-verified
//
#include <hip/hip_runtime.h>

// Cost-volume construction (B=4, C=32, H=64, W=128, D=48).
// out shape (4, 64, 48, 64, 128) fp32:
//   out[b, c   , d, h, w] = (w >= d) ? x[b,c,h,w]   : 0
//   out[b, c+32, d, h, w] = (w >= d) ? y[b,c,h,w-d] : 0
// Pure data movement: ~402.7 MB of write-once output -> NT b128 stores,
// y rows staged in LDS via CDNA5 async global->LDS copy (zero pad of D
// floats makes the shifted masked read a plain LDS load), x rows in regs.

#define B_ 4
#define C_ 32
#define H_ 64
#define W_ 128
#define D_ 48

typedef float v4f __attribute__((ext_vector_type(4)));
typedef int   v4i __attribute__((ext_vector_type(4)));

typedef __attribute__((address_space(1))) v4i* gv4i_p;   // global b128
typedef __attribute__((address_space(3))) v4i* lv4i_p;   // LDS b128

#if defined(__gfx1250__) && __has_builtin(__builtin_amdgcn_global_load_async_to_lds_b128)
#define USE_ASYNC_LDS 1
#else
#define USE_ASYNC_LDS 0
#endif

__global__ __launch_bounds__(256)
void cost_volume_kernel(const float* __restrict__ x,
                        const float* __restrict__ y,
                        float* __restrict__ out) {
    // [0, 48): zero pad, [48, 176): the y row for this (b,c,h)
    __shared__ float yl[D_ + W_];

    const int t    = threadIdx.x;
    const int lane = t & 31;
    const int wv   = t >> 5;            // wave index 0..7 (wave32)
    const int bch  = blockIdx.x;        // = (b*C + c)*H + h
    const int h    = bch & (H_ - 1);
    const int c    = (bch >> 6) & (C_ - 1);
    const int b    = bch >> 11;

    const float* xrow = x + (size_t)bch * W_;
    const float* yrow = y + (size_t)bch * W_;

    // Zero the pad region (makes w < d read 0.0 for the shifted y part).
    if (t < D_) yl[t] = 0.0f;

#if USE_ASYNC_LDS
    // CDNA5 async global->LDS copy: wave 0 DMA's the 512B y row into LDS.
    if (wv == 0) {
        __builtin_amdgcn_global_load_async_to_lds_b128(
            (gv4i_p)(yrow + (lane << 2)),
            (lv4i_p)&yl[D_ + (lane << 2)],
            0, 0);
#if __has_builtin(__builtin_amdgcn_s_wait_asynccnt)
        __builtin_amdgcn_s_wait_asynccnt(0);
#else
        asm volatile("s_wait_asynccnt 0" ::: "memory");
#endif
    }
#else
    if (t >= D_ && t < D_ + W_) yl[t] = yrow[t - D_];
#endif
    __syncthreads();

    const int w0 = lane << 2;
    const v4f xv = *(const v4f*)(xrow + w0);   // x row lives in registers

    const size_t planeC = (size_t)D_ * H_ * W_;   // 393216 floats per (b,c)
    const size_t rowD   = (size_t)H_ * W_;        // 8192 floats per d
    const size_t baseX  = (size_t)(b * (2 * C_) + c) * planeC
                        + (size_t)h * W_ + (size_t)w0;
    const size_t baseY  = baseX + (size_t)C_ * planeC;

    // Each wave covers d = wv, wv+8, ..., wv+40  (6 disparities, all 48 total).
#pragma unroll
    for (int i = 0; i < D_ / 8; ++i) {
        const int d = wv + (i << 3);
        v4f xo, yo;
#pragma unroll
        for (int e = 0; e < 4; ++e) {
            xo[e] = ((w0 + e) >= d) ? xv[e] : 0.0f;
            yo[e] = yl[D_ + w0 + e - d];   // pad zone supplies the zeros
        }
        // Write-once 402 MB stream: non-temporal b128 stores.
        __builtin_nontemporal_store(xo, (v4f*)(out + baseX + (size_t)d * rowD));
        __builtin_nontemporal_store(yo, (v4f*)(out + baseY + (size_t)d * rowD));
    }
}

extern "C" void kernel_launch(void* const* d_in, const int* in_sizes, int n_in,
                              void* d_out, int out_size, void* d_ws, size_t ws_size,
                              hipStream_t stream) {
    const float* x = (const float*)d_in[0];
    const float* y = (const float*)d_in[1];
    float* out = (float*)d_out;

    dim3 grid(B_ * C_ * H_);   // 8192 blocks, one (b,c,h) row pair each
    dim3 block(256);           // 8 wave32s
    cost_volume_kernel<<<grid, block, 0, stream>>>(x, y, out);
}